// PMPGNN_28836410425872
// MI455X (gfx1250) — compile-verified
//
#include <hip/hip_runtime.h>
#include <hip/hip_bf16.h>

// ---------------------------------------------------------------------------
// GATv2 GNN forward for MI455X (gfx1250, wave32).
// Dense GEMMs use native fp32 WMMA (V_WMMA_F32_16X16X4_F32) with the W matrix
// staged through LDS (320KB/WGP) in 64-row chunks, conflict-free padded
// stride, fully unrolled K-chunk so ds_loads use immediate offsets.
// Edge phases: warp-per-edge float4 gather, int-punned float atomic max,
// atomicAdd segment softmax, coalesced atomic scatter.
// ---------------------------------------------------------------------------

typedef __attribute__((ext_vector_type(2))) float v2f;
typedef __attribute__((ext_vector_type(8))) float v8f;

#define N_NODES   16384
#define HID       128
#define OUT_DIM   40
#define ALPHA_C   0.1f
#define NEG_SLOPE 0.2f
#define LN_EPS    1e-5f

// -------------------------------- WMMA GEMM --------------------------------
// C[M,NACT] = act(A[M,K] @ W[K,NACT] + bias). 8 waves/block; each wave owns a
// 16-row strip across NT 16-col tiles (16*NT >= NACT). W staged in LDS per
// 64-row K-chunk. A-frag (16x4 f32): lanes 0-15 hold K=k,k+1; lanes 16-31 hold
// K=k+2,k+3. C/D: vgpr r -> row r (lanes 0-15) / r+8 (lanes 16-31), col=lane%16.
// NOTE: grid must be exactly M/128 blocks (M % 128 == 0) -- barriers in-loop.
template <int NT, int NACT, bool RELU>
__global__ __launch_bounds__(256)
void gemm_wmma_f32(const float* __restrict__ A, const float* __restrict__ W,
                   const float* __restrict__ bias, float* __restrict__ C,
                   int M, int K) {
  constexpr int NBASE   = (NACT + 15) & ~15;                  // == 16*NT
  constexpr int NSTRIDE = (NBASE % 32 == 16) ? NBASE : NBASE + 16; // 144/80/48
  constexpr int KB      = 64;                                 // K-chunk rows
  __shared__ float sW[KB * NSTRIDE];                          // <= 36 KB

  const int tid  = threadIdx.x;
  const int wave = tid >> 5;
  const int lane = tid & 31;
  const int lr   = lane & 15;                 // 0..15
  const int kk   = (lane >> 4) << 1;          // 0 or 2
  const int row0 = (blockIdx.x * 8 + wave) << 4;

  v8f acc[NT] = {};
  const float* arow = A + (size_t)(row0 + lr) * K + kk;

  for (int kb = 0; kb < K; kb += KB) {
    __syncthreads();                          // prev chunk fully consumed
    // cooperative stage of W[kb:kb+KB, :] (zero-padded to NSTRIDE columns)
    for (int i = tid; i < KB * NSTRIDE; i += 256) {
      const int r = i / NSTRIDE;
      const int c = i - r * NSTRIDE;
      sW[i] = (c < NACT) ? W[(size_t)(kb + r) * NACT + c] : 0.0f;
    }
    __syncthreads();

    const float* ap = arow + kb;
    const float* bp = &sW[kk * NSTRIDE + lr];
#pragma unroll
    for (int k0 = 0; k0 < KB; k0 += 8) {
      v2f a0, a1;                             // two k-steps of A (global_load_b64)
      a0.x = ap[k0];     a0.y = ap[k0 + 1];
      a1.x = ap[k0 + 4]; a1.y = ap[k0 + 5];
#pragma unroll
      for (int t = 0; t < NT; ++t) {
        v2f b;
        b.x = bp[k0 * NSTRIDE + t * 16];
        b.y = bp[k0 * NSTRIDE + t * 16 + NSTRIDE];
        acc[t] = __builtin_amdgcn_wmma_f32_16x16x4_f32(
            false, a0, false, b, (short)0, acc[t], false, false);
      }
#pragma unroll
      for (int t = 0; t < NT; ++t) {
        v2f b;
        b.x = bp[(k0 + 4) * NSTRIDE + t * 16];
        b.y = bp[(k0 + 4) * NSTRIDE + t * 16 + NSTRIDE];
        acc[t] = __builtin_amdgcn_wmma_f32_16x16x4_f32(
            false, a1, false, b, (short)0, acc[t], false, false);
      }
    }
  }

  const int rofs = (lane >> 4) << 3;          // 0 or 8
#pragma unroll
  for (int t = 0; t < NT; ++t) {
    const int col = t * 16 + lr;
    const bool ok = (NACT % 16 == 0) || (col < NACT);   // folds for 128/64
    const float bv = ok ? bias[col] : 0.0f;
#pragma unroll
    for (int r = 0; r < 8; ++r) {
      const int row = row0 + r + rofs;
      float v = acc[t][r] + bv;
      if (RELU) v = fmaxf(v, 0.0f);
      if (ok) C[(size_t)row * NACT + col] = v;
    }
  }
}

// ------------------------------ GAT edge phase ------------------------------

__device__ __forceinline__ void atomicMaxF(float* addr, float v) {
  if (v >= 0.0f)
    atomicMax(reinterpret_cast<int*>(addr), __float_as_int(v));
  else
    atomicMin(reinterpret_cast<unsigned int*>(addr), __float_as_uint(v));
}

// smax = -inf, denom = 0, g[N*128] = 0
__global__ __launch_bounds__(256)
void gat_init(float* __restrict__ smax, float* __restrict__ denom,
              float* __restrict__ g, int n) {
  const int i = blockIdx.x * blockDim.x + threadIdx.x;
  if (i < n * HID) g[i] = 0.0f;
  if (i < n) { smax[i] = -__builtin_inff(); denom[i] = 0.0f; }
}

// warp-per-edge: s = att . leaky_relu(xl[src] + xr[dst]); atomic max per dst.
// float4 loads -> coalesced b128 gathers.
__global__ __launch_bounds__(256)
void gat_edge_score(const int* __restrict__ eidx, int E,
                    const float* __restrict__ xl, const float* __restrict__ xr,
                    const float* __restrict__ att,
                    float* __restrict__ sedge, float* __restrict__ smax) {
  const int e = (blockIdx.x * blockDim.x + threadIdx.x) >> 5;
  if (e >= E) return;
  const int lane = threadIdx.x & 31;
  const int src = eidx[e];
  const int dst = eidx[E + e];
  const float4 l = ((const float4*)(xl + (size_t)src * HID))[lane];
  const float4 r = ((const float4*)(xr + (size_t)dst * HID))[lane];
  const float4 w = ((const float4*)att)[lane];
  float vx = l.x + r.x, vy = l.y + r.y, vz = l.z + r.z, vw = l.w + r.w;
  vx = (vx > 0.0f) ? vx : NEG_SLOPE * vx;
  vy = (vy > 0.0f) ? vy : NEG_SLOPE * vy;
  vz = (vz > 0.0f) ? vz : NEG_SLOPE * vz;
  vw = (vw > 0.0f) ? vw : NEG_SLOPE * vw;
  float partial = vx * w.x + vy * w.y + vz * w.z + vw * w.w;
#pragma unroll
  for (int off = 16; off > 0; off >>= 1) partial += __shfl_xor(partial, off, 32);
  if (lane == 0) {
    sedge[e] = partial;
    atomicMaxF(smax + dst, partial);
  }
}

// thread-per-edge: ex = exp(s - smax[dst]); denom[dst] += ex (ex kept in sedge)
__global__ __launch_bounds__(256)
void gat_edge_exp(const int* __restrict__ eidx, int E,
                  const float* __restrict__ smax,
                  float* __restrict__ sedge, float* __restrict__ denom) {
  const int e = blockIdx.x * blockDim.x + threadIdx.x;
  if (e >= E) return;
  const int dst = eidx[E + e];
  const float ex = expf(sedge[e] - smax[dst]);
  sedge[e] = ex;
  atomicAdd(denom + dst, ex);
}

// warp-per-edge: g[dst] += xl[src] * (ex / denom[dst])
__global__ __launch_bounds__(256)
void gat_edge_scatter(const int* __restrict__ eidx, int E,
                      const float* __restrict__ xl,
                      const float* __restrict__ sedge,
                      const float* __restrict__ denom, float* __restrict__ g) {
  const int e = (blockIdx.x * blockDim.x + threadIdx.x) >> 5;
  if (e >= E) return;
  const int lane = threadIdx.x & 31;
  const int src = eidx[e];
  const int dst = eidx[E + e];
  const float alpha = sedge[e] / denom[dst];
  const float4 l = ((const float4*)(xl + (size_t)src * HID))[lane];
  float* pg = g + (size_t)dst * HID + lane * 4;
  atomicAdd(pg + 0, l.x * alpha);
  atomicAdd(pg + 1, l.y * alpha);
  atomicAdd(pg + 2, l.z * alpha);
  atomicAdd(pg + 3, l.w * alpha);
}

// ------------------- fused concat+scale+bias(+res)+LN+ReLU -------------------
__global__ __launch_bounds__(256)
void block_finalize(const float* __restrict__ g1, const float* __restrict__ g2,
                    const float* __restrict__ bias_a, const float* __restrict__ bias_b,
                    const float* __restrict__ ng, const float* __restrict__ nb,
                    const float* __restrict__ residual, float* __restrict__ out,
                    int n) {
  const int row = (blockIdx.x * blockDim.x + threadIdx.x) >> 5;
  if (row >= n) return;
  const int lane = threadIdx.x & 31;
  float v[8];
  float s = 0.0f, sq = 0.0f;
#pragma unroll
  for (int j = 0; j < 8; ++j) {
    const int c = lane + 32 * j;
    float val;
    if (c < HID) val = (1.0f - ALPHA_C) * (g1[(size_t)row * HID + c] + bias_a[c]);
    else         val = ALPHA_C * (g2[(size_t)row * HID + (c - HID)] + bias_b[c - HID]);
    if (residual) val += residual[(size_t)row * (2 * HID) + c];
    v[j] = val;
    s  += val;
    sq += val * val;
  }
#pragma unroll
  for (int off = 16; off > 0; off >>= 1) {
    s  += __shfl_xor(s, off, 32);
    sq += __shfl_xor(sq, off, 32);
  }
  const float mean = s * (1.0f / (2 * HID));
  const float var  = sq * (1.0f / (2 * HID)) - mean * mean;
  const float inv  = rsqrtf(var + LN_EPS);
#pragma unroll
  for (int j = 0; j < 8; ++j) {
    const int c = lane + 32 * j;
    float o = (v[j] - mean) * inv * ng[c] + nb[c];
    out[(size_t)row * (2 * HID) + c] = fmaxf(o, 0.0f);
  }
}

// ------------------------------- log_softmax --------------------------------
__global__ __launch_bounds__(256)
void logsoftmax40(const float* __restrict__ z, float* __restrict__ out, int n) {
  const int row = (blockIdx.x * blockDim.x + threadIdx.x) >> 5;
  if (row >= n) return;
  const int lane = threadIdx.x & 31;
  const float* p = z + (size_t)row * OUT_DIM;
  const float v0 = p[lane];                                    // lane < 32 < 40
  const bool  h1 = (lane + 32) < OUT_DIM;
  const float v1 = h1 ? p[lane + 32] : -__builtin_inff();
  float m = fmaxf(v0, v1);
#pragma unroll
  for (int off = 16; off > 0; off >>= 1) m = fmaxf(m, __shfl_xor(m, off, 32));
  float s = expf(v0 - m) + (h1 ? expf(v1 - m) : 0.0f);
#pragma unroll
  for (int off = 16; off > 0; off >>= 1) s += __shfl_xor(s, off, 32);
  const float ls = m + logf(s);
  out[(size_t)row * OUT_DIM + lane] = v0 - ls;
  if (h1) out[(size_t)row * OUT_DIM + lane + 32] = v1 - ls;
}

// ------------------------------- GAT driver ---------------------------------
static void run_gat(const float* h, int K,
                    const float* Wl, const float* bl,
                    const float* Wr, const float* br, const float* att,
                    const int* e, int E,
                    float* xl, float* xr, float* smax, float* denom,
                    float* sedge, float* g, hipStream_t stream) {
  const int N = N_NODES;
  const int gblk = N / 128;                             // 128 rows per block
  gemm_wmma_f32<8, HID, false><<<gblk, 256, 0, stream>>>(h, Wl, bl, xl, N, K);
  gemm_wmma_f32<8, HID, false><<<gblk, 256, 0, stream>>>(h, Wr, br, xr, N, K);
  gat_init<<<(N * HID) / 256, 256, 0, stream>>>(smax, denom, g, N);
  const int eb = (E * 32 + 255) / 256;
  gat_edge_score<<<eb, 256, 0, stream>>>(e, E, xl, xr, att, sedge, smax);
  gat_edge_exp<<<(E + 255) / 256, 256, 0, stream>>>(e, E, smax, sedge, denom);
  gat_edge_scatter<<<eb, 256, 0, stream>>>(e, E, xl, sedge, denom, g);
}

// --------------------------------- launch -----------------------------------
extern "C" void kernel_launch(void* const* d_in, const int* in_sizes, int n_in,
                              void* d_out, int out_size, void* d_ws, size_t ws_size,
                              hipStream_t stream) {
  // setup_inputs() dict insertion order, recursively flattened:
  // 0: x | 1 proj_W, 2 proj_b | blk0.a: 3 Wl, 4 bl, 5 Wr, 6 br, 7 att, 8 bias
  // blk0.b: 9..14 | blk1.a: 15..20 | blk1.b: 21..26 | 27 norm_g, 28 norm_b
  // 29 mlp_W1, 30 mlp_b1, 31 mlp_W2, 32 mlp_b2 | 33 e1[2,E1] | 34 e2[2,E2]
  const float* x      = (const float*)d_in[0];
  const float* proj_W = (const float*)d_in[1];
  const float* proj_b = (const float*)d_in[2];
  const float* g0a[6] = {(const float*)d_in[3],  (const float*)d_in[4],
                         (const float*)d_in[5],  (const float*)d_in[6],
                         (const float*)d_in[7],  (const float*)d_in[8]};
  const float* g0b[6] = {(const float*)d_in[9],  (const float*)d_in[10],
                         (const float*)d_in[11], (const float*)d_in[12],
                         (const float*)d_in[13], (const float*)d_in[14]};
  const float* g1a[6] = {(const float*)d_in[15], (const float*)d_in[16],
                         (const float*)d_in[17], (const float*)d_in[18],
                         (const float*)d_in[19], (const float*)d_in[20]};
  const float* g1b[6] = {(const float*)d_in[21], (const float*)d_in[22],
                         (const float*)d_in[23], (const float*)d_in[24],
                         (const float*)d_in[25], (const float*)d_in[26]};
  const float* norm_g = (const float*)d_in[27];
  const float* norm_b = (const float*)d_in[28];
  const float* mlp_W1 = (const float*)d_in[29];
  const float* mlp_b1 = (const float*)d_in[30];
  const float* mlp_W2 = (const float*)d_in[31];
  const float* mlp_b2 = (const float*)d_in[32];
  const int* e1 = (const int*)d_in[33];
  const int* e2 = (const int*)d_in[34];
  const int E1 = in_sizes[33] / 2;
  const int E2 = in_sizes[34] / 2;

  const int N = N_NODES;
  float* ws = (float*)d_ws;
  // scratch (floats): 9*N*128 + 2N + max(E1,E2)  (~79 MB)
  float* h0    = ws;                         // N*128
  float* xl    = h0 + (size_t)N * HID;       // N*128
  float* xr    = xl + (size_t)N * HID;       // N*128
  float* gg1   = xr + (size_t)N * HID;       // N*128
  float* gg2   = gg1 + (size_t)N * HID;      // N*128
  float* hA    = gg2 + (size_t)N * HID;      // N*256
  float* hB    = hA + (size_t)N * 2 * HID;   // N*256
  float* smax  = hB + (size_t)N * 2 * HID;   // N
  float* denom = smax + N;                   // N
  float* sedge = denom + N;                  // max(E1,E2)
  float* z1 = xl;                            // reuse after GAT phases (N*64)
  float* z2 = xr;                            // reuse (N*40)

  const int gblk = N / 128;

  // x -> relu(x @ proj_W + b)                        [N,128]
  gemm_wmma_f32<8, HID, true><<<gblk, 256, 0, stream>>>(x, proj_W, proj_b, h0,
                                                        N, 256);
  // block0 (no residual)
  run_gat(h0, HID, g0a[0], g0a[1], g0a[2], g0a[3], g0a[4], e1, E1,
          xl, xr, smax, denom, sedge, gg1, stream);
  run_gat(h0, HID, g0b[0], g0b[1], g0b[2], g0b[3], g0b[4], e2, E2,
          xl, xr, smax, denom, sedge, gg2, stream);
  block_finalize<<<(N * 32) / 256, 256, 0, stream>>>(
      gg1, gg2, g0a[5], g0b[5], norm_g, norm_b, nullptr, hA, N);

  // block1 (residual = hA)
  run_gat(hA, 2 * HID, g1a[0], g1a[1], g1a[2], g1a[3], g1a[4], e1, E1,
          xl, xr, smax, denom, sedge, gg1, stream);
  run_gat(hA, 2 * HID, g1b[0], g1b[1], g1b[2], g1b[3], g1b[4], e2, E2,
          xl, xr, smax, denom, sedge, gg2, stream);
  block_finalize<<<(N * 32) / 256, 256, 0, stream>>>(
      gg1, gg2, g1a[5], g1b[5], norm_g, norm_b, hA, hB, N);

  // MLP head + log_softmax
  gemm_wmma_f32<4, 64, true><<<gblk, 256, 0, stream>>>(hB, mlp_W1, mlp_b1, z1,
                                                       N, 2 * HID);
  gemm_wmma_f32<3, OUT_DIM, false><<<gblk, 256, 0, stream>>>(z1, mlp_W2, mlp_b2,
                                                             z2, N, 64);
  logsoftmax40<<<(N * 32) / 256, 256, 0, stream>>>(z2, (float*)d_out, N);
}